// Model_8830452760989
// MI455X (gfx1250) — compile-verified
//
#include <hip/hip_runtime.h>

// ---------------------------------------------------------------------------
// MI455X (gfx1250) implementation of the detection post-process reference.
//
// Roofline: the dominant cost is streaming x (261*320*320*4B = 106.9 MB) once
// from HBM: ~4.6us at 23.3 TB/s. The class GEMM (M=102400,N=16,K=256) has
// AI ~8 flop/byte -> memory bound, so we keep full f32 precision and use
// V_WMMA_F32_16X16X4_F32 (exact vs the reference) instead of converting to
// f16. The serial NMS scan is done by ONE wave32 using a precomputed
// 4096x4096 IoU bitmask (popcount + shuffle reduction per row).
// ---------------------------------------------------------------------------

#define NP        102400      // 320*320 pixels
#define FEAT_OFF  5
#define KF        256         // feature channels
#define NCLS      16
#define KTOP      4096
#define THRV      0.5f
#define IOU_THRV  0.2f
#define LDS_W_STRIDE 258      // padded row stride (floats) -> no LDS bank conflicts

typedef float v2f __attribute__((ext_vector_type(2)));
typedef float v8f __attribute__((ext_vector_type(8)));

__device__ __forceinline__ unsigned orderKey(float s) {
  unsigned u = __float_as_uint(s);
  return (u & 0x80000000u) ? ~u : (u | 0x80000000u);
}
__device__ __forceinline__ float keyToFloat(unsigned k) {
  return (k & 0x80000000u) ? __uint_as_float(k & 0x7fffffffu)
                           : __uint_as_float(~k);
}

// ---------------------------------------------------------------------------
// Kernel 1: fused WMMA GEMM (feat @ W^T), sigmoid, max/argmax, score.
// One wave handles a 16-pixel (M) tile; N=16 classes is one WMMA tile;
// K=256 consumed 4 at a time via V_WMMA_F32_16X16X4_F32.
// A layout (16x4 f32): lanes 0-15 hold M=lane, K={0,1} in v0/v1;
//                      lanes 16-31 hold M=lane-16, K={2,3}.
// B layout mirrors A with N in place of M. C/D: lanes 0-15 N=lane, M=r;
// lanes 16-31 N=lane-16, M=r+8.
// ---------------------------------------------------------------------------
__global__ __launch_bounds__(256) void score_gemm_kernel(
    const float* __restrict__ x, const float* __restrict__ W,
    float* __restrict__ score, int* __restrict__ clsidx)
{
  __shared__ float wlds[NCLS * LDS_W_STRIDE];
  const int t = threadIdx.x;
  for (int i = t; i < NCLS * KF; i += 256)
    wlds[(i >> 8) * LDS_W_STRIDE + (i & 255)] = W[i];
  __syncthreads();

  const int wave  = t >> 5;
  const int lane  = t & 31;
  const int half  = lane >> 4;     // 0: K-pair {0,1}, 1: K-pair {2,3}
  const int lr    = lane & 15;
  const int p_base = (blockIdx.x * 8 + wave) * 16;   // 800 blocks * 8 waves * 16 = NP

  const float* aptr = x + (size_t)(FEAT_OFF + 2 * half) * NP + p_base + lr;
  const float* bptr = &wlds[lr * LDS_W_STRIDE + 2 * half];

  v8f acc = {};
  #pragma unroll 8
  for (int k = 0; k < KF; k += 4) {
    v2f a, b;
    a.x = aptr[(size_t)k * NP];
    a.y = aptr[(size_t)(k + 1) * NP];
    float2 bv = *reinterpret_cast<const float2*>(bptr + k);  // ds_load_b64
    b.x = bv.x; b.y = bv.y;
    acc = __builtin_amdgcn_wmma_f32_16x16x4_f32(
        /*neg_a=*/false, a, /*neg_b=*/false, b,
        /*c_mod=*/(short)0, acc, /*reuse_a=*/false, /*reuse_b=*/false);
  }

  // acc[r]: lanes 0-15 -> logits[M=r][N=lane]; lanes 16-31 -> logits[M=r+8][N=lane-16]
  #pragma unroll
  for (int r = 0; r < 8; ++r) {
    float v  = acc[r];
    int  idx = lr;
    #pragma unroll
    for (int off = 8; off >= 1; off >>= 1) {
      float ov = __shfl_xor(v, off, 16);
      int   oi = __shfl_xor(idx, off, 16);
      if (ov > v || (ov == v && oi < idx)) { v = ov; idx = oi; }  // argmax ties: low index
    }
    if (lr == 0) {
      int pix = p_base + r + half * 8;
      float s0  = x[pix];                       // channel 0
      float sig = 1.0f / (1.0f + __expf(-s0));
      score[pix]  = (sig > THRV) ? sig * v : -__builtin_inff();
      clsidx[pix] = idx;
    }
  }
}

// ---------------------------------------------------------------------------
// Top-K = exact two-level radix select + one-block bitonic sort.
// ---------------------------------------------------------------------------
__global__ void zero_kernel(unsigned* hist1, unsigned* hist2, unsigned* meta) {
  int i = blockIdx.x * blockDim.x + threadIdx.x;
  if (i < 65536) { hist1[i] = 0; hist2[i] = 0; }
  if (i < 16) meta[i] = 0;
}

__global__ void hist1_kernel(const float* __restrict__ score, unsigned* __restrict__ hist) {
  int i = blockIdx.x * blockDim.x + threadIdx.x;
  atomicAdd(&hist[orderKey(score[i]) >> 16], 1u);
}

__global__ __launch_bounds__(1024) void findT1_kernel(const unsigned* __restrict__ hist,
                                                      unsigned* __restrict__ meta) {
  __shared__ unsigned csum[1024];
  int t = threadIdx.x;
  unsigned s = 0;
  for (int b = 0; b < 64; ++b) s += hist[t * 64 + b];
  csum[t] = s;
  __syncthreads();
  if (t == 0) {
    unsigned acc = 0; int chunk = 0;
    for (int c = 1023; c >= 0; --c) {
      if (acc + csum[c] >= KTOP) { chunk = c; break; }
      acc += csum[c];
    }
    unsigned T1 = 0, cntA = acc;
    for (int b = chunk * 64 + 63; b >= chunk * 64; --b) {
      unsigned h = hist[b];
      if (cntA + h >= KTOP) { T1 = (unsigned)b; break; }
      cntA += h;
    }
    meta[0] = T1; meta[1] = cntA;
  }
}

__global__ void hist2_kernel(const float* __restrict__ score,
                             const unsigned* __restrict__ meta,
                             unsigned* __restrict__ hist) {
  int i = blockIdx.x * blockDim.x + threadIdx.x;
  unsigned u = orderKey(score[i]);
  if ((u >> 16) == meta[0]) atomicAdd(&hist[u & 0xffffu], 1u);
}

__global__ __launch_bounds__(1024) void findT2_kernel(const unsigned* __restrict__ hist2,
                                                      unsigned* __restrict__ meta) {
  __shared__ unsigned csum[1024];
  int t = threadIdx.x;
  unsigned s = 0;
  for (int b = 0; b < 64; ++b) s += hist2[t * 64 + b];
  csum[t] = s;
  __syncthreads();
  if (t == 0) {
    unsigned T1 = meta[0];
    unsigned acc = meta[1];            // count strictly above bin T1
    int chunk = 0;
    for (int c = 1023; c >= 0; --c) {
      if (acc + csum[c] >= KTOP) { chunk = c; break; }
      acc += csum[c];
    }
    unsigned T2 = 0, cntGT = acc;
    for (int b = chunk * 64 + 63; b >= chunk * 64; --b) {
      unsigned h = hist2[b];
      if (cntGT + h >= KTOP) { T2 = (unsigned)b; break; }
      cntGT += h;
    }
    meta[2] = (T1 << 16) | T2;   // exact 32-bit threshold key
    meta[3] = cntGT;             // # elements with key > threshold (< K)
    meta[4] = KTOP - cntGT;      // # needed from the key == threshold set
  }
}

__global__ void compact_kernel(const float* __restrict__ score,
                               unsigned* __restrict__ meta,
                               unsigned long long* __restrict__ keybuf,
                               unsigned* __restrict__ eqbuf) {
  int i = blockIdx.x * blockDim.x + threadIdx.x;
  unsigned UT = meta[2];
  unsigned u = orderKey(score[i]);
  if (u > UT) {
    unsigned pos = atomicAdd(&meta[5], 1u);          // pos < cntGT by construction
    keybuf[pos] = ((unsigned long long)u << 32) | (unsigned)(~i);
  } else if (u == UT) {
    unsigned pos = atomicAdd(&meta[6], 1u);
    if (pos < KTOP) eqbuf[pos] = (unsigned)i;
  }
}

__device__ void bitonicU32(unsigned* s, int n, int tid, int nthr) {  // ascending
  for (int k = 2; k <= n; k <<= 1)
    for (int j = k >> 1; j > 0; j >>= 1) {
      __syncthreads();
      for (int i = tid; i < n; i += nthr) {
        int p = i ^ j;
        if (p > i) {
          bool up = ((i & k) == 0);
          unsigned A = s[i], B = s[p];
          if ((A > B) == up) { s[i] = B; s[p] = A; }
        }
      }
    }
  __syncthreads();
}
__device__ void bitonicU64(unsigned long long* s, int n, int tid, int nthr) {  // ascending
  for (int k = 2; k <= n; k <<= 1)
    for (int j = k >> 1; j > 0; j >>= 1) {
      __syncthreads();
      for (int i = tid; i < n; i += nthr) {
        int p = i ^ j;
        if (p > i) {
          bool up = ((i & k) == 0);
          unsigned long long A = s[i], B = s[p];
          if ((A > B) == up) { s[i] = B; s[p] = A; }
        }
      }
    }
  __syncthreads();
}

__global__ __launch_bounds__(1024) void select_sort_kernel(
    const unsigned long long* __restrict__ keybuf,
    const unsigned* __restrict__ eqbuf,
    const unsigned* __restrict__ meta,
    unsigned long long* __restrict__ skeys)
{
  __shared__ unsigned long long sk[KTOP];     // 32 KB LDS
  unsigned* se = (unsigned*)sk;
  int tid = threadIdx.x;
  unsigned cntGT = meta[3], UT = meta[2];
  unsigned eqCnt = meta[6]; if (eqCnt > KTOP) eqCnt = KTOP;

  // 1) sort the "== threshold" indices ascending; smallest indices win ties
  for (int i = tid; i < KTOP; i += 1024)
    se[i] = (i < (int)eqCnt) ? eqbuf[i] : 0xffffffffu;
  bitonicU32(se, KTOP, tid, 1024);

  // 2) stage the exact K keys (cntGT strict winners + need tie winners)
  unsigned long long tmp[4];
  #pragma unroll
  for (int c = 0; c < 4; ++c) {
    int i = tid + c * 1024;
    if (i < (int)cntGT) tmp[c] = keybuf[i];
    else {
      unsigned e = se[i - (int)cntGT];
      tmp[c] = ((unsigned long long)UT << 32) | (unsigned)(~e);
    }
  }
  __syncthreads();
  #pragma unroll
  for (int c = 0; c < 4; ++c) sk[tid + c * 1024] = tmp[c];

  // 3) full sort; emit descending (score desc, index asc on ties)
  bitonicU64(sk, KTOP, tid, 1024);
  for (int i = tid; i < KTOP; i += 1024) skeys[i] = sk[KTOP - 1 - i];
}

// ---------------------------------------------------------------------------
// Gather boxes/cls/area for the sorted top-K; build alive bitmask via ballot.
// ---------------------------------------------------------------------------
__global__ void gather_kernel(const unsigned long long* __restrict__ skeys,
                              const float* __restrict__ x,
                              const int* __restrict__ clsidx,
                              float* __restrict__ topv, float* __restrict__ boxes,
                              float* __restrict__ area, int* __restrict__ clsk,
                              unsigned* __restrict__ aliveWords)
{
  int k = blockIdx.x * blockDim.x + threadIdx.x;
  unsigned long long key = skeys[k];
  unsigned u   = (unsigned)(key >> 32);
  unsigned idx = ~(unsigned)key;
  float tv = keyToFloat(u);
  if (idx >= NP) { idx = 0; tv = -__builtin_inff(); }
  float l = x[(size_t)1 * NP + idx];
  float t = x[(size_t)2 * NP + idx];
  float r = x[(size_t)3 * NP + idx];
  float b = x[(size_t)4 * NP + idx];
  topv[k] = tv;
  ((float4*)boxes)[k] = make_float4(l, t, r, b);
  area[k] = (r - l) * (b - t);
  clsk[k] = clsidx[idx];
  unsigned long long m = __ballot(tv != -__builtin_inff());
  if ((threadIdx.x & 31) == 0) aliveWords[k >> 5] = (unsigned)m;
}

// ---------------------------------------------------------------------------
// IoU >= 0.2 bitmask: row i, word w covers boxes [32w, 32w+32).
// ---------------------------------------------------------------------------
__global__ __launch_bounds__(128) void iou_bits_kernel(
    const float* __restrict__ boxes, const float* __restrict__ area,
    unsigned* __restrict__ iouBits)
{
  int i = blockIdx.x;
  int w = threadIdx.x;
  const float4* bx = (const float4*)boxes;
  float4 bi = bx[i];
  float ai = area[i];
  unsigned bits = 0;
  int j0 = w * 32;
  for (int b = 0; b < 32; ++b) {
    float4 bj = bx[j0 + b];
    float iw = fminf(bi.z, bj.z) - fmaxf(bi.x, bj.x);
    float ih = fminf(bi.w, bj.w) - fmaxf(bi.y, bj.y);
    iw = fmaxf(iw, 0.0f); ih = fmaxf(ih, 0.0f);
    float inter = iw * ih;
    float aj = (bj.z - bj.x) * (bj.w - bj.y);
    float uni = ai + aj - inter + 1e-10f;
    if (inter >= IOU_THRV * uni) bits |= (1u << b);   // inter/uni >= thr, uni > 0
  }
  iouBits[i * 128 + w] = bits;
}

// ---------------------------------------------------------------------------
// Serial NMS scan on a single wave32. alive/kept = 4096 bits = 4 regs/lane.
// Lane l owns words [4l, 4l+4). Next IoU row is prefetched (double buffer).
// ---------------------------------------------------------------------------
__global__ __launch_bounds__(32) void nms_kernel(
    const unsigned* __restrict__ iouBits, const unsigned* __restrict__ aliveWords,
    const float* __restrict__ area, unsigned* __restrict__ keptWords)
{
  int lane = threadIdx.x;
  unsigned a0 = aliveWords[4 * lane + 0];
  unsigned a1 = aliveWords[4 * lane + 1];
  unsigned a2 = aliveWords[4 * lane + 2];
  unsigned a3 = aliveWords[4 * lane + 3];
  unsigned k0 = 0, k1 = 0, k2 = 0, k3 = 0;
  const uint4* rows = (const uint4*)iouBits;   // 32 uint4 per row
  uint4 cur = rows[lane];
  for (int i = 0; i < KTOP; ++i) {
    uint4 nxt = (i + 1 < KTOP) ? rows[(i + 1) * 32 + lane] : make_uint4(0, 0, 0, 0);
    int wi = i >> 5, bi = i & 31;
    int owner = wi >> 2, sub = wi & 3;
    unsigned mybit = (sub == 0) ? a0 : (sub == 1) ? a1 : (sub == 2) ? a2 : a3;
    unsigned wv = __shfl(mybit, owner);
    bool is_cur = (wv >> bi) & 1u;
    if (lane == owner) {                       // alive[i] = false (before counting)
      unsigned m = ~(1u << bi);
      if (sub == 0) a0 &= m; else if (sub == 1) a1 &= m;
      else if (sub == 2) a2 &= m; else a3 &= m;
    }
    int c = __popc(cur.x & a0) + __popc(cur.y & a1) +
            __popc(cur.z & a2) + __popc(cur.w & a3);
    c += __shfl_xor(c, 16);
    c += __shfl_xor(c, 8);
    c += __shfl_xor(c, 4);
    c += __shfl_xor(c, 2);
    c += __shfl_xor(c, 1);
    float ar = area[i];
    bool accept = is_cur && (ar >= 4.0f) && (c >= 10);
    if (accept) {
      a0 &= ~cur.x; a1 &= ~cur.y; a2 &= ~cur.z; a3 &= ~cur.w;  // suppress overlaps
      if (lane == owner) {
        if (sub == 0) k0 |= (1u << bi); else if (sub == 1) k1 |= (1u << bi);
        else if (sub == 2) k2 |= (1u << bi); else k3 |= (1u << bi);
      }
    }
    cur = nxt;
  }
  keptWords[4 * lane + 0] = k0;
  keptWords[4 * lane + 1] = k1;
  keptWords[4 * lane + 2] = k2;
  keptWords[4 * lane + 3] = k3;
}

// ---------------------------------------------------------------------------
// Final transform: de-letterbox, clamp, emit (K,7) dets.
// ---------------------------------------------------------------------------
__global__ void finalize_kernel(const unsigned* __restrict__ keptWords,
                                const float* __restrict__ topv,
                                const float* __restrict__ boxes,
                                const int* __restrict__ clsk,
                                const float* __restrict__ padding,
                                const float* __restrict__ ratio,
                                const int* __restrict__ imw,
                                const int* __restrict__ imh,
                                float* __restrict__ dets)
{
  int k = blockIdx.x * blockDim.x + threadIdx.x;
  bool kept = (keptWords[k >> 5] >> (k & 31)) & 1u;
  float tv = topv[k];
  bool keep = kept && (tv >= THRV);
  float inv_r = 1.0f / ratio[0];
  float px = padding[0], py = padding[1];
  float Wf = (float)imw[0], Hf = (float)imh[0];
  float4 b = ((const float4*)boxes)[k];
  float x1 = fmaxf((b.x - px) * inv_r, 0.0f);
  float y1 = fmaxf((b.y - py) * inv_r, 0.0f);
  float x2 = fminf((b.z - px) * inv_r, Wf);
  float y2 = fminf((b.w - py) * inv_r, Hf);
  keep = keep && (x1 < Wf) && (y1 < Hf);
  float* d = dets + (size_t)k * 7;
  if (keep) {
    d[0] = 1.0f; d[1] = tv; d[2] = x1; d[3] = y1;
    d[4] = x2;   d[5] = y2; d[6] = (float)clsk[k];
  } else {
    #pragma unroll
    for (int q = 0; q < 7; ++q) d[q] = 0.0f;
  }
}

// ---------------------------------------------------------------------------
extern "C" void kernel_launch(void* const* d_in, const int* in_sizes, int n_in,
                              void* d_out, int out_size, void* d_ws, size_t ws_size,
                              hipStream_t stream) {
  (void)in_sizes; (void)n_in; (void)out_size; (void)ws_size;
  const float* x       = (const float*)d_in[0];
  const float* W       = (const float*)d_in[1];
  const float* padding = (const float*)d_in[2];
  const float* ratio   = (const float*)d_in[3];
  const int*   imw     = (const int*)d_in[4];
  const int*   imh     = (const int*)d_in[5];
  float* dets = (float*)d_out;

  char* ws = (char*)d_ws;                                   // total ~3.5 MB
  float*              score   = (float*)(ws + 0);           // 409600 B
  int*                clsidx  = (int*)(ws + 409600);        // 409600 B
  unsigned*           hist1   = (unsigned*)(ws + 819200);   // 256 KB
  unsigned*           hist2   = (unsigned*)(ws + 1081344);  // 256 KB
  unsigned*           meta    = (unsigned*)(ws + 1343488);  // 64 B
  unsigned long long* keybuf  = (unsigned long long*)(ws + 1343552); // 32 KB
  unsigned*           eqbuf   = (unsigned*)(ws + 1376320);  // 16 KB
  unsigned long long* skeys   = (unsigned long long*)(ws + 1392704); // 32 KB
  float*              topv    = (float*)(ws + 1425472);     // 16 KB
  float*              boxes   = (float*)(ws + 1441856);     // 64 KB
  float*              area    = (float*)(ws + 1507392);     // 16 KB
  int*                clsk    = (int*)(ws + 1523776);       // 16 KB
  unsigned*           aliveW  = (unsigned*)(ws + 1540160);  // 512 B
  unsigned*           keptW   = (unsigned*)(ws + 1540672);  // 512 B
  unsigned*           iouBits = (unsigned*)(ws + 1541184);  // 2 MB

  zero_kernel       <<<256, 256, 0, stream>>>(hist1, hist2, meta);
  score_gemm_kernel <<<800, 256, 0, stream>>>(x, W, score, clsidx);
  hist1_kernel      <<<400, 256, 0, stream>>>(score, hist1);
  findT1_kernel     <<<1, 1024, 0, stream>>>(hist1, meta);
  hist2_kernel      <<<400, 256, 0, stream>>>(score, meta, hist2);
  findT2_kernel     <<<1, 1024, 0, stream>>>(hist2, meta);
  compact_kernel    <<<400, 256, 0, stream>>>(score, meta, keybuf, eqbuf);
  select_sort_kernel<<<1, 1024, 0, stream>>>(keybuf, eqbuf, meta, skeys);
  gather_kernel     <<<16, 256, 0, stream>>>(skeys, x, clsidx, topv, boxes, area, clsk, aliveW);
  iou_bits_kernel   <<<4096, 128, 0, stream>>>(boxes, area, iouBits);
  nms_kernel        <<<1, 32, 0, stream>>>(iouBits, aliveW, area, keptW);
  finalize_kernel   <<<16, 256, 0, stream>>>(keptW, topv, boxes, clsk, padding, ratio, imw, imh, dets);
}